// SpectrumHead_7911329759543
// MI455X (gfx1250) — compile-verified
//
#include <hip/hip_runtime.h>
#include <hip/hip_bf16.h>
#include <math.h>

// ---------------------------------------------------------------------------
// SpectrumHead on MI455X (gfx1250): rfft2 as two WMMA-f32 GEMM stages against
// L2-resident twiddle matrices; double-buffered async global->LDS tile DMA;
// fused magnitude/log1p/histogram epilogue.
// ---------------------------------------------------------------------------

typedef float v2f __attribute__((ext_vector_type(2)));
typedef float v8f __attribute__((ext_vector_type(8)));
typedef int   v4i_ __attribute__((__vector_size__(16)));

#define HDIM   512
#define W2     257
#define NPAD   272            // 257 padded to 17*16
#define NBATCH 64
#define NCH    3
#define NIMG   (NBATCH * NCH) // 192
#define MROWS  (NIMG * HDIM)  // 98304
#define PAD    20             // LDS row stride (floats): 16B-aligned rows,
                              // conflict-free (row*20 mod 64 distinct, 16 rows)

// workspace layout (in floats)
#define OFF_CW  ((size_t)0)
#define SZ_CW   ((size_t)HDIM * NPAD)       // 139264
#define OFF_SW  (OFF_CW + SZ_CW)
#define OFF_CH  (OFF_SW + SZ_CW)
#define SZ_CH   ((size_t)HDIM * HDIM)       // 262144
#define OFF_SH  (OFF_CH + SZ_CH)
#define OFF_HB  (OFF_SH + SZ_CH)
#define SZ_HB   ((size_t)NBATCH * 24)
#define OFF_YR  (OFF_HB + SZ_HB)
#define SZ_Y    ((size_t)MROWS * NPAD)      // 26.7M floats
#define OFF_YI  (OFF_YR + SZ_Y)
// total ~54.3M floats ~ 217 MB of d_ws

#if __has_builtin(__builtin_amdgcn_global_load_async_to_lds_b128)
#define HAVE_ASYNC 1
#else
#define HAVE_ASYNC 0
#endif

#define AS1 __attribute__((address_space(1)))
#define AS3 __attribute__((address_space(3)))

__device__ __forceinline__ v8f wmma_f32(v2f a, v2f b, v8f c) {
  // V_WMMA_F32_16X16X4_F32 : D(16x16,f32) = A(16x4,f32) * B(4x16,f32) + C
  return __builtin_amdgcn_wmma_f32_16x16x4_f32(
      false, a, false, b, (short)0, c, false, false);
}

// 16-byte global->LDS tile copy: async DMA (ASYNCcnt) when available,
// else synchronous float4 staging through VGPRs.
__device__ __forceinline__ void copy16(const float* __restrict__ g,
                                       float* __restrict__ l) {
#if HAVE_ASYNC
  float* gnc = const_cast<float*>(g);
  __builtin_amdgcn_global_load_async_to_lds_b128(
      (AS1 v4i_*)gnc, (AS3 v4i_*)l, 0, 0);
#else
  *reinterpret_cast<float4*>(l) = *reinterpret_cast<const float4*>(g);
#endif
}

__device__ __forceinline__ void wait_async() {
#if HAVE_ASYNC
  asm volatile("s_wait_asynccnt 0" ::: "memory");
#endif
}

// ---------------------------------------------------------------------------
// Twiddle generation + histogram-buffer zeroing (runs every launch; exact
// integer phase reduction (n*k) mod 512 keeps cosf/sinf accurate).
// ---------------------------------------------------------------------------
__global__ void init_kernel(float* __restrict__ ws) {
  const int t = blockIdx.x * blockDim.x + threadIdx.x;
  const float step = 6.28318530717958647692f / 512.0f;
  if (t < HDIM * NPAD) {
    const int w = t / NPAD, j = t - w * NPAD;
    const int ph = (w * j) & 511;
    const float ang = step * (float)ph;
    ws[OFF_CW + t] = cosf(ang);
    ws[OFF_SW + t] = -sinf(ang);   // Yi = X * (-sin)  -> true imaginary part
  }
  if (t < HDIM * HDIM) {
    const int k = t >> 9, h = t & 511;
    const int ph = (k * h) & 511;
    const float ang = step * (float)ph;
    ws[OFF_CH + t] = cosf(ang);
    ws[OFF_SH + t] = sinf(ang);
  }
  if (t < NBATCH * 24) ws[OFF_HB + t] = 0.0f;
}

// ---------------------------------------------------------------------------
// Stage 1: row DFT as one big GEMM.
//   Yr[m, j] = sum_w X[m, w] * cos(2*pi*w*j/512)
//   Yi[m, j] = sum_w X[m, w] * -sin(2*pi*w*j/512)
// M = 98304, K = 512, N = 272.  WG = 128 threads (4 waves), tile 256x16.
// Double-buffered LDS: chunk k+1 DMA overlaps chunk k WMMA stream.
// ---------------------------------------------------------------------------
__global__ __launch_bounds__(128) void stage1_rowdft(const float* __restrict__ X,
                                                     float* __restrict__ ws) {
  const float* __restrict__ Cw = ws + OFF_CW;
  const float* __restrict__ Sw = ws + OFF_SW;
  float* __restrict__ Yr = ws + OFF_YR;
  float* __restrict__ Yi = ws + OFF_YI;

  __shared__ float As[2][256 * PAD];
  __shared__ float Bc[2][16 * PAD];
  __shared__ float Bs[2][16 * PAD];

  const int tid     = threadIdx.x;
  const int lane    = tid & 31;
  const int wid     = tid >> 5;
  const int n       = lane & 15;
  const int koff    = (lane >> 4) * 2;     // A/B fragment K offset per lane half
  const int rowhalf = (lane >> 4) * 8;     // D fragment M offset per lane half
  const size_t m0   = (size_t)blockIdx.y * 256;
  const int j0      = blockIdx.x * 16;

  // tile-load decomposition (issued by every thread each chunk)
  const int ar  = tid >> 2;                // A rows handled by this thread
  const int ac4 = tid & 3;                 // (with +32-row steps below)
  const int bsel = tid >> 6;               // 0: Bc, 1: Bs
  const int bidx = tid & 63;
  const int br_ = bidx >> 2, bc4 = bidx & 3;

  v8f accR[4], accI[4];
#pragma unroll
  for (int mt = 0; mt < 4; ++mt)
#pragma unroll
    for (int e = 0; e < 8; ++e) { accR[mt][e] = 0.0f; accI[mt][e] = 0.0f; }

  // issue loads for a chunk into buffer `b`
  auto load_tiles = [&](int b, int kk) {
#pragma unroll
    for (int i = 0; i < 8; ++i) {
      const int r = ar + i * 32;
      copy16(X + (m0 + r) * HDIM + kk + ac4 * 4, &As[b][r * PAD + ac4 * 4]);
    }
    const float* bsrc = (bsel == 0)
        ? (Cw + (size_t)(kk + br_) * NPAD + j0 + bc4 * 4)
        : (Sw + (size_t)(kk + br_) * NPAD + j0 + bc4 * 4);
    float* bdst = (bsel == 0) ? &Bc[b][br_ * PAD + bc4 * 4]
                              : &Bs[b][br_ * PAD + bc4 * 4];
    copy16(bsrc, bdst);
  };

  load_tiles(0, 0);
  int buf = 0;
  for (int kk = 0; kk < HDIM; kk += 16) {
    wait_async();        // own async loads into `buf` complete
    __syncthreads();     // all waves' loads visible; prev compute on buf^1 done
    if (kk + 16 < HDIM) load_tiles(buf ^ 1, kk + 16);
#pragma unroll
    for (int s = 0; s < 4; ++s) {          // 4 k-steps of 4
      const int ks = s * 4;
      v2f bc, bs;
      bc[0] = Bc[buf][(ks + koff) * PAD + n];
      bc[1] = Bc[buf][(ks + koff + 1) * PAD + n];
      bs[0] = Bs[buf][(ks + koff) * PAD + n];
      bs[1] = Bs[buf][(ks + koff + 1) * PAD + n];
#pragma unroll
      for (int mt = 0; mt < 4; ++mt) {
        const int rowL = wid * 64 + mt * 16 + (lane & 15);
        v2f a;
        a[0] = As[buf][rowL * PAD + ks + koff];
        a[1] = As[buf][rowL * PAD + ks + koff + 1];
        accR[mt] = wmma_f32(a, bc, accR[mt]);
        accI[mt] = wmma_f32(a, bs, accI[mt]);
      }
    }
    buf ^= 1;
  }

#pragma unroll
  for (int mt = 0; mt < 4; ++mt)
#pragma unroll
    for (int e = 0; e < 8; ++e) {
      const size_t gm = m0 + wid * 64 + mt * 16 + rowhalf + e;
      const size_t o = gm * NPAD + j0 + n;
      Yr[o] = accR[mt][e];
      Yi[o] = accI[mt][e];
    }
}

// ---------------------------------------------------------------------------
// Stage 2: column DFT per image (A = shared 512x512 cos/sin twiddles,
// B = Yr/Yi of image), fused |Z|, channel mean, log1p and binning.
//   Zr = Ch*Yr + Sh*Yi ;  Zi = Ch*Yi - Sh*Yr   (magnitude is sign-invariant)
// grid (17 j-tiles, 2 k-tiles, 64 batches), 128 threads, tile 256x16.
// ---------------------------------------------------------------------------
__global__ __launch_bounds__(128) void stage2_coldft_hist(float* __restrict__ ws) {
  const float* __restrict__ Ch = ws + OFF_CH;
  const float* __restrict__ Sh = ws + OFF_SH;
  const float* __restrict__ Yr = ws + OFF_YR;
  const float* __restrict__ Yi = ws + OFF_YI;
  float* __restrict__ hbuf = ws + OFF_HB;

  __shared__ float Ac[2][256 * PAD];
  __shared__ float Asn[2][256 * PAD];
  __shared__ float Br[2][16 * PAD];
  __shared__ float Bi[2][16 * PAD];
  __shared__ float bins[24];

  const int tid     = threadIdx.x;
  const int lane    = tid & 31;
  const int wid     = tid >> 5;
  const int n       = lane & 15;
  const int koff    = (lane >> 4) * 2;
  const int rowhalf = (lane >> 4) * 8;
  const int b       = blockIdx.z;
  const int m0      = blockIdx.y * 256;
  const int j0      = blockIdx.x * 16;

  if (tid < 24) bins[tid] = 0.0f;

  const int ar  = tid >> 2;
  const int ac4 = tid & 3;
  const int bsel = tid >> 6;
  const int bidx = tid & 63;
  const int br_ = bidx >> 2, bc4 = bidx & 3;

  float csum[4][8];
#pragma unroll
  for (int mt = 0; mt < 4; ++mt)
#pragma unroll
    for (int e = 0; e < 8; ++e) csum[mt][e] = 0.0f;

  for (int c = 0; c < NCH; ++c) {
    const float* __restrict__ YrB = Yr + (size_t)(b * NCH + c) * HDIM * NPAD;
    const float* __restrict__ YiB = Yi + (size_t)(b * NCH + c) * HDIM * NPAD;

    auto load_tiles = [&](int bb, int kk) {
#pragma unroll
      for (int i = 0; i < 8; ++i) {
        const int r = ar + i * 32;
        copy16(Ch + (size_t)(m0 + r) * HDIM + kk + ac4 * 4,
               &Ac[bb][r * PAD + ac4 * 4]);
        copy16(Sh + (size_t)(m0 + r) * HDIM + kk + ac4 * 4,
               &Asn[bb][r * PAD + ac4 * 4]);
      }
      const float* bsrc = (bsel == 0)
          ? (YrB + (size_t)(kk + br_) * NPAD + j0 + bc4 * 4)
          : (YiB + (size_t)(kk + br_) * NPAD + j0 + bc4 * 4);
      float* bdst = (bsel == 0) ? &Br[bb][br_ * PAD + bc4 * 4]
                                : &Bi[bb][br_ * PAD + bc4 * 4];
      copy16(bsrc, bdst);
    };

    v8f accR[4], accI[4];
#pragma unroll
    for (int mt = 0; mt < 4; ++mt)
#pragma unroll
      for (int e = 0; e < 8; ++e) { accR[mt][e] = 0.0f; accI[mt][e] = 0.0f; }

    load_tiles(0, 0);
    int buf = 0;
    for (int kk = 0; kk < HDIM; kk += 16) {
      wait_async();
      __syncthreads();
      if (kk + 16 < HDIM) load_tiles(buf ^ 1, kk + 16);
#pragma unroll
      for (int s = 0; s < 4; ++s) {
        const int ks = s * 4;
        v2f brf, bif, brn;
        brf[0] = Br[buf][(ks + koff) * PAD + n];
        brf[1] = Br[buf][(ks + koff + 1) * PAD + n];
        bif[0] = Bi[buf][(ks + koff) * PAD + n];
        bif[1] = Bi[buf][(ks + koff + 1) * PAD + n];
        brn[0] = -brf[0];
        brn[1] = -brf[1];
#pragma unroll
        for (int mt = 0; mt < 4; ++mt) {
          const int rowL = wid * 64 + mt * 16 + (lane & 15);
          v2f acf, asf;
          acf[0] = Ac[buf][rowL * PAD + ks + koff];
          acf[1] = Ac[buf][rowL * PAD + ks + koff + 1];
          asf[0] = Asn[buf][rowL * PAD + ks + koff];
          asf[1] = Asn[buf][rowL * PAD + ks + koff + 1];
          accR[mt] = wmma_f32(acf, brf, accR[mt]);   // + Ch*Yr
          accR[mt] = wmma_f32(asf, bif, accR[mt]);   // + Sh*Yi
          accI[mt] = wmma_f32(acf, bif, accI[mt]);   // + Ch*Yi
          accI[mt] = wmma_f32(asf, brn, accI[mt]);   // - Sh*Yr
        }
      }
      buf ^= 1;
    }
    __syncthreads();   // buffers reused next channel iteration

    const float orthoscale = 1.0f / 512.0f;  // 1/sqrt(512*512)
#pragma unroll
    for (int mt = 0; mt < 4; ++mt)
#pragma unroll
      for (int e = 0; e < 8; ++e)
        csum[mt][e] += sqrtf(accR[mt][e] * accR[mt][e] +
                             accI[mt][e] * accI[mt][e]) * orthoscale;
  }

  // fused histogram epilogue: bins on (k, j) frequency coordinates
  const float inv3 = 1.0f / 3.0f;
  const int j = j0 + n;
#pragma unroll
  for (int mt = 0; mt < 4; ++mt)
#pragma unroll
    for (int e = 0; e < 8; ++e) {
      if (j < W2) {
        const int kf = m0 + wid * 64 + mt * 16 + rowhalf + e;
        const float yy = -1.0f + 2.0f * (float)kf / 511.0f;
        const float xx = (float)j / 256.0f;
        float rr = sqrtf(yy * yy + xx * xx);
        rr = fminf(rr, 1.0f - 1e-8f);
        const float th = atan2f(yy, xx + 1e-9f) + 1.57079632679489662f;
        int rb = (int)(rr * 16.0f);
        rb = rb < 0 ? 0 : (rb > 15 ? 15 : rb);
        int ob = (int)(th * (8.0f / 3.14159265358979323846f));
        ob = ob < 0 ? 0 : (ob > 7 ? 7 : ob);
        const float m = log1pf(csum[mt][e] * inv3);
        atomicAdd(&bins[rb], m);
        atomicAdd(&bins[16 + ob], m);
      }
    }
  __syncthreads();
  if (tid < 24) atomicAdd(&hbuf[b * 24 + tid], bins[tid]);
}

// ---------------------------------------------------------------------------
// Stage 3: normalize histograms, project h @ W.T + b  ->  out [64, 64]
// ---------------------------------------------------------------------------
__global__ void project_kernel(const float* __restrict__ ws,
                               const float* __restrict__ Wm,
                               const float* __restrict__ bias,
                               float* __restrict__ out) {
  const int b = blockIdx.x;
  const int p = threadIdx.x;
  const float* h = ws + OFF_HB + (size_t)b * 24;
  float rs = 0.0f, as = 0.0f;
#pragma unroll
  for (int q = 0; q < 16; ++q) rs += h[q];
#pragma unroll
  for (int q = 16; q < 24; ++q) as += h[q];
  rs = 1.0f / (rs + 1e-6f);
  as = 1.0f / (as + 1e-6f);
  float acc = bias[p];
#pragma unroll
  for (int q = 0; q < 16; ++q) acc += h[q] * rs * Wm[p * 24 + q];
#pragma unroll
  for (int q = 16; q < 24; ++q) acc += h[q] * as * Wm[p * 24 + q];
  out[b * 64 + p] = acc;
}

// ---------------------------------------------------------------------------
extern "C" void kernel_launch(void* const* d_in, const int* in_sizes, int n_in,
                              void* d_out, int out_size, void* d_ws, size_t ws_size,
                              hipStream_t stream) {
  (void)in_sizes; (void)n_in; (void)out_size; (void)ws_size;
  const float* X    = (const float*)d_in[0];   // [64,3,512,512]
  const float* Wm   = (const float*)d_in[1];   // [64,24]
  const float* bias = (const float*)d_in[2];   // [64]
  float* out = (float*)d_out;                  // [64,64]
  float* ws  = (float*)d_ws;

  init_kernel<<<1024, 256, 0, stream>>>(ws);
  stage1_rowdft<<<dim3(17, MROWS / 256), 128, 0, stream>>>(X, ws);
  stage2_coldft_hist<<<dim3(17, 2, NBATCH), 128, 0, stream>>>(ws);
  project_kernel<<<NBATCH, 64, 0, stream>>>(ws, Wm, bias, out);
}